// RegularizedWassersteinDistance_824633721696
// MI455X (gfx1250) — compile-verified
//
#include <hip/hip_runtime.h>

// ---------------------------------------------------------------------------
// Sinkhorn / regularized Wasserstein on MI455X (gfx1250).
//   n = 4096, batch = 64, 50 iterations, K symmetric (cost is symmetric).
//   GEMM: out[b,n] = sum_k in[b,k] * K[k,n]   via V_WMMA_F32_16X16X4_F32.
// K (64 MB fp32) is L2-resident (192 MB global L2); each GEMM pass reads K
// exactly once (disjoint 128-col groups per workgroup). The small A operand
// (u/v, 1 MB) is staged into LDS with GLOBAL_LOAD_ASYNC_TO_LDS_B128 (CDNA5
// async engine, ASYNCcnt) in a [b][k] pitch-68 layout so each WMMA A operand
// is a single conflict-free ds_load_b64. Split-K x8 for occupancy,
// deterministic 2-stage reduction.
// ---------------------------------------------------------------------------

typedef __attribute__((ext_vector_type(2))) float v2f;
typedef __attribute__((ext_vector_type(8))) float v8f;

#define NDIM    4096             // n = 64*64
#define BATCH   64
#define NB      (BATCH * NDIM)   // 262144 elements per [b,n] matrix
#define KCHUNK  512              // split-K chunk per workgroup (8 chunks)
#define KBLK    64               // k-block staged in LDS at a time
#define LPITCH  68               // LDS row pitch in floats ([b][k] layout)
#define NCHUNKS 8

// ---------------------------------------------------------------------------
// init u = v = 1/n
// ---------------------------------------------------------------------------
__global__ __launch_bounds__(256) void init_uv(float* __restrict__ u,
                                               float* __restrict__ v) {
  int i = blockIdx.x * 256 + threadIdx.x;
  if (i < NB) {
    const float c = 1.0f / (float)NDIM;
    u[i] = c;
    v[i] = c;
  }
}

// ---------------------------------------------------------------------------
// Split-K GEMM:  partial[kc][b][n] = sum_{k in chunk kc} vin[b,k] * B[k,n]
// where B = Kmat (MULC=0) or Kmat.*Cmat (MULC=1, final distance pass only).
// Grid: (32 n-groups of 128 cols, 8 k-chunks). Block: 256 threads = 8 waves.
// Each wave: one 16-col N tile, all 4 M(batch) tiles, K-loop of 512.
// ---------------------------------------------------------------------------
template <int MULC>
__global__ __launch_bounds__(256) void sinkhorn_gemm(
    const float* __restrict__ vin,   // [64][4096]
    const float* __restrict__ Kmat,  // [4096][4096]
    const float* __restrict__ Cmat,  // [4096][4096]
    float* __restrict__ partial) {   // [8][64][4096]
  __shared__ float vs[BATCH * LPITCH];  // staged vin block, layout [b][k]

  const int lane = threadIdx.x & 31;
  const int wave = threadIdx.x >> 5;
  const int row  = lane & 15;   // M (and N) index within tile
  const int koff = lane >> 4;   // selects K pair {0,1} vs {2,3}
  const int n0   = blockIdx.x * 128 + wave * 16;
  const int col  = n0 + row;
  const int kbase = blockIdx.y * KCHUNK;

  // B-operand base pointers: rows (kbase + 2*koff + kl), column col.
  // Advanced by KBLK rows per k-block so row offsets are immediates.
  const float* Bp = Kmat + (size_t)(kbase + 2 * koff) * NDIM + col;
  const float* Cp = Cmat + (size_t)(kbase + 2 * koff) * NDIM + col;

  // Per-thread async-stage addresses: thread (b = tid&63, q = tid>>6) copies
  // 4x b128 covering vin[b, kbase+kb+q*16 : +16] into vs[b][q*16 : +16].
  const int sb = threadIdx.x & 63;
  const int sq = threadIdx.x >> 6;
  const float* gsrc0 = vin + (size_t)sb * NDIM + kbase + sq * 16;
  const unsigned ldst0 =
      (unsigned)(size_t)(const void*)(&vs[sb * LPITCH + sq * 16]);

  v8f acc[4];
#pragma unroll
  for (int m = 0; m < 4; ++m) acc[m] = (v8f){};

  for (int kb = 0; kb < KCHUNK; kb += KBLK) {
    __syncthreads();  // previous block fully consumed before overwrite
    // ---- async stage: global -> LDS, 4 x B128 per thread ----
    {
      const float* gsrc = gsrc0 + kb;
#pragma unroll
      for (int i = 0; i < 4; ++i) {
        asm volatile("global_load_async_to_lds_b128 %0, %1, off"
                     :
                     : "v"(ldst0 + (unsigned)(i * 16)), "v"(gsrc + i * 4)
                     : "memory");
      }
      asm volatile("s_wait_asynccnt 0x0" ::: "memory");
    }
    __syncthreads();

#pragma unroll
    for (int kl = 0; kl < KBLK; kl += 4) {
      // ---- B operand: rows kl, kl+1 at column col (immediate offsets) ----
      float b0 = Bp[kl * NDIM];
      float b1 = Bp[kl * NDIM + NDIM];
      if (MULC) {
        b0 *= Cp[kl * NDIM];
        b1 *= Cp[kl * NDIM + NDIM];
      }
      v2f bop;
      bop[0] = b0;
      bop[1] = b1;

      // ---- A operands: one ds_load_b64 each, vs[m*16+row][k0 : k0+1] ----
      const int k0 = kl + 2 * koff;
#pragma unroll
      for (int m = 0; m < 4; ++m) {
        v2f aop = *(const v2f*)(&vs[(m * 16 + row) * LPITCH + k0]);
        acc[m] = __builtin_amdgcn_wmma_f32_16x16x4_f32(
            /*neg_a=*/false, aop, /*neg_b=*/false, bop,
            /*c_mod=*/(short)0, acc[m], /*reuse_a=*/false, /*reuse_b=*/false);
      }
    }
    Bp += KBLK * NDIM;
    if (MULC) Cp += KBLK * NDIM;
  }

  // Write partials. C/D layout: VGPR r -> M = r + 8*koff, N = col.
  float* pout = partial + (size_t)blockIdx.y * NB + col;
#pragma unroll
  for (int m = 0; m < 4; ++m) {
#pragma unroll
    for (int r = 0; r < 8; ++r) {
      const int brow = m * 16 + koff * 8 + r;
      pout[brow * NDIM] = acc[m][r];
    }
  }
}

// ---------------------------------------------------------------------------
// Deterministic split-K reduction (+ optional Sinkhorn divide):
//   s = sum_kc partial[kc][i];  out[i] = doDiv ? numer[i]/(s+1e-8) : s
// ---------------------------------------------------------------------------
__global__ __launch_bounds__(256) void reduce_div(
    const float* __restrict__ partial, const float* __restrict__ numer,
    float* __restrict__ outv, int doDiv) {
  const int i = blockIdx.x * 256 + threadIdx.x;
  if (i >= NB) return;
  float s = 0.0f;
#pragma unroll
  for (int kc = 0; kc < NCHUNKS; ++kc) s += partial[(size_t)kc * NB + i];
  outv[i] = doDiv ? (numer[i] / (s + 1e-8f)) : s;
}

// ---------------------------------------------------------------------------
// Deterministic final dot:  out[0] = sum_i u[i] * W[i]   (single workgroup)
// ---------------------------------------------------------------------------
__global__ __launch_bounds__(1024) void dot_reduce(const float* __restrict__ u,
                                                   const float* __restrict__ W,
                                                   float* __restrict__ out) {
  __shared__ float red[1024];
  float s = 0.0f;
  for (int i = threadIdx.x; i < NB; i += 1024) s += u[i] * W[i];
  red[threadIdx.x] = s;
  __syncthreads();
  for (int off = 512; off > 0; off >>= 1) {
    if ((int)threadIdx.x < off) red[threadIdx.x] += red[threadIdx.x + off];
    __syncthreads();
  }
  if (threadIdx.x == 0) out[0] = red[0];
}

// ---------------------------------------------------------------------------
// Host launcher: 1 init + 50 * (gemm, div, gemm, div) + final gemm/reduce/dot
// All launches stream-ordered; graph-capture safe (no allocs / syncs).
// ---------------------------------------------------------------------------
extern "C" void kernel_launch(void* const* d_in, const int* in_sizes, int n_in,
                              void* d_out, int out_size, void* d_ws,
                              size_t ws_size, hipStream_t stream) {
  const float* x    = (const float*)d_in[0];  // [64,64,64] == [64,4096]
  const float* y    = (const float*)d_in[1];
  const float* cost = (const float*)d_in[2];  // [4096,4096]
  const float* Kmat = (const float*)d_in[3];  // [4096,4096], symmetric

  float* u       = (float*)d_ws;          //  1 MB
  float* v       = u + NB;                //  1 MB
  float* W       = v + NB;                //  1 MB
  float* partial = W + NB;                //  8 MB  (total 11 MB scratch)

  const dim3 ggrid(32, NCHUNKS);  // 256 workgroups, 8 waves each
  const int rgrid = NB / 256;     // 1024 blocks

  init_uv<<<rgrid, 256, 0, stream>>>(u, v);

  for (int it = 0; it < 50; ++it) {
    // nu = x / (v @ K + 1e-8)        (K symmetric => v @ K^T == v @ K)
    sinkhorn_gemm<0><<<ggrid, 256, 0, stream>>>(v, Kmat, cost, partial);
    reduce_div<<<rgrid, 256, 0, stream>>>(partial, x, u, 1);
    // nv = y / (u @ K + 1e-8)
    sinkhorn_gemm<0><<<ggrid, 256, 0, stream>>>(u, Kmat, cost, partial);
    reduce_div<<<rgrid, 256, 0, stream>>>(partial, y, v, 1);
  }

  // W[b,i] = sum_j (K*C)[i,j] v[b,j]  (K*C symmetric); dist = sum u .* W
  sinkhorn_gemm<1><<<ggrid, 256, 0, stream>>>(v, Kmat, cost, partial);
  reduce_div<<<rgrid, 256, 0, stream>>>(partial, x, W, 0);
  dot_reduce<<<1, 1024, 0, stream>>>(u, W, (float*)d_out);
}